// BandLinear_14474039788006
// MI455X (gfx1250) — compile-verified
//
#include <hip/hip_runtime.h>
#include <stdint.h>

typedef __attribute__((ext_vector_type(2))) float v2f;
typedef __attribute__((ext_vector_type(4))) float v4f;
typedef __attribute__((ext_vector_type(8))) float v8f;

#define IN_F   4096
#define OUT_F  4096
#define BW     8
#define N_TOK  16384

#define OTILES 8                 // output tiles per block (one per wave)
#define OBLK   (OTILES * 16)     // 128 outputs per block
#define SLICE  (OBLK + 2 * BW)   // 144 input columns incl. band halo
#define TTILES 4                 // token tiles per wave
#define TBLK   (TTILES * 16)     // 64 tokens per block
#define XS_STRIDE 148            // 16B-aligned rows; 148 mod 64 = 20 = 4*5,
                                 // 5 coprime 16 -> 16 row reads hit distinct
                                 // bank quads; both lane halves together
                                 // cover all 64 banks conflict-free
#define QPR (SLICE / 4)          // 36 float4 quads per row

__global__ __launch_bounds__(256) void band_linear_wmma_f32(
    const float* __restrict__ x,
    const float* __restrict__ weight,
    const float* __restrict__ bias,
    float* __restrict__ out)
{
    // shared x slice: 64 tokens x 144 inputs (padded rows)
    __shared__ __align__(16) float sX[TBLK * XS_STRIDE];
    // per-wave masked W^T tile (K=32 x N=16), packed as K-pairs:
    // sWt[w][(k/2)*16 + n] is float2 {Wt[k][n], Wt[k+1][n]}
    __shared__ __align__(16) float sWt[OTILES][32 * 16];

    const int tid  = threadIdx.x;
    const int wv   = tid >> 5;
    const int lane = tid & 31;
    const int m    = lane & 15;     // M for A-frag, N for B/C/D frags
    const int h    = lane >> 4;     // lane half

    const int obb      = blockIdx.x;            // output super-tile (0..31)
    const int o_base   = obb * OBLK + wv * 16;  // this wave's 16 outputs
    const int is_base  = obb * OBLK - BW;       // shared slice start (may be <0)
    const int tok_base = blockIdx.y * TBLK;

    // ---- stage x slice into LDS with async b128 global->LDS copies ----
    // halo = 8 = two full quads, so every 16B quad is entirely in- or
    // out-of-range: OOB quads are zero-filled, in-range quads go async.
    {
        const uint64_t xbase = (uint64_t)(uintptr_t)x;
        #pragma unroll
        for (int it = 0; it < (TBLK * QPR) / 256; ++it) {     // 9 iters
            const int q  = it * 256 + tid;
            const int r  = q / QPR;
            const int c  = (q - r * QPR) * 4;
            const int i0 = is_base + c;                        // multiple of 4
            float* ldsp  = &sX[r * XS_STRIDE + c];
            if ((unsigned)i0 < (unsigned)IN_F) {
                const unsigned lds_off = (unsigned)(uintptr_t)ldsp;
                const int voff = ((tok_base + r) * IN_F + i0) * 4;
                asm volatile("global_load_async_to_lds_b128 %0, %1, %2"
                             :: "v"(lds_off), "v"(voff), "s"(xbase)
                             : "memory");
            } else {
                *(v4f*)ldsp = (v4f)0.0f;   // band halo outside [0, IN_F)
            }
        }
    }

    // ---- each wave stages its masked W^T tile (weights are L2-resident) ----
    {
        const int ibw = o_base - BW;
        for (int idx = lane; idx < 512; idx += 32) {
            const int k = idx >> 4;
            const int n = idx & 15;
            const int i = ibw + k;
            const int o = o_base + n;
            const int d = o - i;
            float v = 0.0f;
            if ((unsigned)i < (unsigned)IN_F && d >= -BW && d <= BW)
                v = weight[(size_t)o * IN_F + i];
            sWt[wv][((k >> 1) * 16 + n) * 2 + (k & 1)] = v;
        }
    }

    // drain this wave's async copies, then publish the tile to all waves
    asm volatile("s_wait_asynccnt 0x0" ::: "memory");
    __syncthreads();

    const float bv = bias[o_base + m];

    // ---- 4 token tiles x 8 WMMA steps (K=32 band slice each) ----
    #pragma unroll
    for (int tt = 0; tt < TTILES; ++tt) {
        v8f c;
        #pragma unroll
        for (int j = 0; j < 8; ++j) c[j] = bv;

        const float* xrow = &sX[(tt * 16 + m) * XS_STRIDE + wv * 16];
        #pragma unroll
        for (int s = 0; s < 8; ++s) {
            v2f a = *(const v2f*)(xrow + 4 * s + 2 * h);                 // {x[M][4s+2h], x[M][4s+2h+1]}
            v2f b = *(const v2f*)(&sWt[wv][((2 * s + h) * 16 + m) * 2]); // {Wt[4s+2h][N], Wt[4s+2h+1][N]}
            c = __builtin_amdgcn_wmma_f32_16x16x4_f32(
                    false, a, false, b, (short)0, c, false, false);
        }

        // D layout: VGPR r -> (token = tok_base + tt*16 + r + 8h, out = o_base + N)
        // pure streaming output: non-temporal stores keep band weights in L2
        const size_t row0 = (size_t)(tok_base + tt * 16 + 8 * h) * OUT_F
                          + (size_t)(o_base + m);
        #pragma unroll
        for (int r = 0; r < 8; ++r)
            __builtin_nontemporal_store(c[r], &out[row0 + (size_t)r * OUT_F]);
    }
}

extern "C" void kernel_launch(void* const* d_in, const int* in_sizes, int n_in,
                              void* d_out, int out_size, void* d_ws, size_t ws_size,
                              hipStream_t stream) {
    const float* x    = (const float*)d_in[0];
    const float* w    = (const float*)d_in[1];
    const float* bias = (const float*)d_in[2];
    // d_in[3] (mask) not needed: the band is applied analytically.
    float* out = (float*)d_out;

    dim3 grid(OUT_F / OBLK, N_TOK / TBLK);   // 32 x 256 blocks
    band_linear_wmma_f32<<<grid, 256, 0, stream>>>(x, w, bias, out);
}